// MinHashEncoder_27590869909669
// MI455X (gfx1250) — compile-verified
//
#include <hip/hip_runtime.h>
#include <hip/hip_bf16.h>

// Problem constants (from the reference).
#define HH 768       // NUM_HASHES
#define VV 30000     // VOCAB
#define BB 64        // batch
#define SS 1024      // sequence
#define SCHUNK 128   // tokens per gather block
#define NCHUNK (SS / SCHUNK)

typedef float v2f __attribute__((ext_vector_type(2)));
typedef float v8f __attribute__((ext_vector_type(8)));

// ---------------------------------------------------------------------------
// Kernel 0: init signature buffer to +inf (int bits, for int atomicMin).
// ---------------------------------------------------------------------------
__global__ void mh_init_inf(int* __restrict__ out, int n) {
    int i = blockIdx.x * blockDim.x + threadIdx.x;
    if (i < n) out[i] = 0x7f800000;  // +inf
}

// ---------------------------------------------------------------------------
// Kernel 1: transpose hashed_vocab (H,V) row-major -> (V,H) row-major in d_ws.
// 32x32 LDS tiles, padded stride to kill bank conflicts. 184 MB HBM traffic
// total (~8us at 23.3 TB/s) -- one-time cost that converts every later gather
// into a contiguous 3KB row read.
// ---------------------------------------------------------------------------
__global__ __launch_bounds__(256) void mh_transpose(const float* __restrict__ A,
                                                    float* __restrict__ T) {
    __shared__ float tile[32][33];
    const int tx = threadIdx.x, ty = threadIdx.y;   // block (32,8)
    const int v0 = blockIdx.x * 32;
    const int h0 = blockIdx.y * 32;                 // H=768 divides by 32
#pragma unroll
    for (int j = 0; j < 4; ++j) {
        int v = v0 + tx;
        int h = h0 + ty + j * 8;
        if (v < VV) tile[ty + j * 8][tx] = A[(size_t)h * VV + v];
    }
    __syncthreads();
#pragma unroll
    for (int j = 0; j < 4; ++j) {
        int v = v0 + ty + j * 8;
        int h = h0 + tx;
        if (v < VV) T[(size_t)v * HH + h] = tile[tx][ty + j * 8];
    }
}

// ---------------------------------------------------------------------------
// Kernel 2: gather + min using the transposed (V,H) table.
// grid = (64 batches, 8 token-chunks); 192 threads, one float4 (4 hashes) per
// thread -> one coalesced 3KB row read per token, float4 (b128) loads, L2-hot.
// Partial mins merged via int atomicMin (valid: all values are >= 0 floats).
// ---------------------------------------------------------------------------
__global__ __launch_bounds__(192) void mh_gather_min(const int* __restrict__ ids,
                                                     const int* __restrict__ mask,
                                                     const float* __restrict__ T,
                                                     float* __restrict__ out) {
    const int b   = blockIdx.x;
    const int c   = blockIdx.y;
    const int tid = threadIdx.x;

    __shared__ int s_ids[SCHUNK];
    const int base = b * SS + c * SCHUNK;
    if (tid < SCHUNK) {
        int id = ids[base + tid];
        int m  = mask[base + tid];
        s_ids[tid] = (m == 1 && id > 100 && id < VV) ? id : -1;
    }
    __syncthreads();

    const int h0 = tid * 4;  // 192 threads * 4 = 768 hashes
    const float inf = __builtin_inff();
    float4 mv = make_float4(inf, inf, inf, inf);

    for (int t = 0; t < SCHUNK; ++t) {
        // Prefetch next token's row segment (emits global_prefetch_b8).
        if (t + 1 < SCHUNK) {
            int nid = s_ids[t + 1];
            if (nid >= 0) __builtin_prefetch(&T[(size_t)nid * HH + h0], 0, 0);
        }
        int id = s_ids[t];
        if (id >= 0) {
            const float4 v = *reinterpret_cast<const float4*>(&T[(size_t)id * HH + h0]);
            mv.x = fminf(mv.x, v.x);
            mv.y = fminf(mv.y, v.y);
            mv.z = fminf(mv.z, v.z);
            mv.w = fminf(mv.w, v.w);
        }
    }

    int* o = reinterpret_cast<int*>(&out[(size_t)b * HH + h0]);
    atomicMin(&o[0], __float_as_int(mv.x));
    atomicMin(&o[1], __float_as_int(mv.y));
    atomicMin(&o[2], __float_as_int(mv.z));
    atomicMin(&o[3], __float_as_int(mv.w));
}

// ---------------------------------------------------------------------------
// Kernel 2b: fallback if d_ws can't hold the 92MB transposed table.
// Direct strided gather from (H,V); 256 threads x 3 hashes each.
// ---------------------------------------------------------------------------
__global__ __launch_bounds__(256) void mh_gather_min_direct(const int* __restrict__ ids,
                                                            const int* __restrict__ mask,
                                                            const float* __restrict__ HV,
                                                            float* __restrict__ out) {
    const int b   = blockIdx.x;
    const int c   = blockIdx.y;
    const int tid = threadIdx.x;

    __shared__ int s_ids[SCHUNK];
    const int base = b * SS + c * SCHUNK;
    if (tid < SCHUNK) {
        int id = ids[base + tid];
        int m  = mask[base + tid];
        s_ids[tid] = (m == 1 && id > 100 && id < VV) ? id : -1;
    }
    __syncthreads();

    const float inf = __builtin_inff();
    float mv[3] = {inf, inf, inf};
    for (int t = 0; t < SCHUNK; ++t) {
        int id = s_ids[t];
        if (id >= 0) {
#pragma unroll
            for (int j = 0; j < 3; ++j) {
                int h = tid + 256 * j;
                mv[j] = fminf(mv[j], HV[(size_t)h * VV + id]);
            }
        }
    }
#pragma unroll
    for (int j = 0; j < 3; ++j) {
        int h = tid + 256 * j;
        atomicMin(reinterpret_cast<int*>(&out[(size_t)b * HH + h]),
                  __float_as_int(mv[j]));
    }
}

// ---------------------------------------------------------------------------
// Kernel 3: L2-normalize rows of the (64,768) signature matrix.
// Sum-of-squares per row = diagonal of X * X^T, computed on the matrix pipe
// with V_WMMA_F32_16X16X4_F32. For this symmetric case the A (16x4, lane=M,
// vgpr=K-pair) and B (4x16 = X^T, row-striped across lanes) register layouts
// coincide, so one loaded v2f serves as both operands. One wave per 16-row
// tile, 192 chained WMMAs over K=768. EXEC is all-ones (32 threads, no
// divergence before the WMMA loop).
// ---------------------------------------------------------------------------
__global__ __launch_bounds__(32) void mh_normalize_wmma(float* __restrict__ out) {
    const int tile = blockIdx.x;        // 4 tiles of 16 rows
    const int lane = threadIdx.x;       // wave32
    const int row  = lane & 15;         // M index
    const int koff = (lane >> 4) * 2;   // K sub-offset within a K=4 chunk

    float* base = out + (size_t)(tile * 16) * HH;

    v8f c = {0.f, 0.f, 0.f, 0.f, 0.f, 0.f, 0.f, 0.f};
    for (int k0 = 0; k0 < HH; k0 += 4) {
        v2f a;
        a.x = base[(size_t)row * HH + k0 + koff];
        a.y = base[(size_t)row * HH + k0 + koff + 1];
        // D = A * A^T + C : diagonal accumulates row sum-of-squares.
        c = __builtin_amdgcn_wmma_f32_16x16x4_f32(
                /*neg_a=*/false, a, /*neg_b=*/false, a,
                /*c_mod=*/(short)0, c, /*reuse_a=*/false, /*reuse_b=*/false);
    }

    // Spill D to LDS and pull the diagonal (C/D layout: VGPR r holds rows r
    // in lanes 0-15 and r+8 in lanes 16-31, N = lane%16).
    __shared__ float lds[32 * 8];
    __shared__ float s_inv[16];
#pragma unroll
    for (int r = 0; r < 8; ++r) lds[lane * 8 + r] = c[r];
    __syncthreads();

    if (lane < 16) {
        const int m = lane;
        float ss = (m < 8) ? lds[m * 8 + m]                 // VGPR m, lane m
                           : lds[(m + 16) * 8 + (m - 8)];   // VGPR m-8, lane m+16
        float nrm = sqrtf(ss);
        s_inv[m] = 1.0f / fmaxf(nrm, 1e-12f);               // matches F.normalize eps
    }
    __syncthreads();

    for (int idx = lane; idx < 16 * HH; idx += 32) {
        int r = idx / HH, col = idx % HH;
        base[(size_t)r * HH + col] *= s_inv[r];
    }
}

// ---------------------------------------------------------------------------
// Launch
// ---------------------------------------------------------------------------
extern "C" void kernel_launch(void* const* d_in, const int* in_sizes, int n_in,
                              void* d_out, int out_size, void* d_ws, size_t ws_size,
                              hipStream_t stream) {
    const int*   ids  = (const int*)d_in[0];     // (64,1024) int32
    const int*   mask = (const int*)d_in[1];     // (64,1024) int32
    const float* hv   = (const float*)d_in[2];   // (768,30000) fp32
    float*       out  = (float*)d_out;           // (64,768) fp32

    // 0) signatures <- +inf
    mh_init_inf<<<(BB * HH + 255) / 256, 256, 0, stream>>>((int*)d_out, BB * HH);

    const size_t need = (size_t)VV * HH * sizeof(float);  // 92.16 MB
    if (ws_size >= need) {
        // 1) transpose (H,V) -> (V,H) in scratch (one-time 184MB HBM traffic)
        float* T = (float*)d_ws;
        dim3 tb(32, 8);
        dim3 tg((VV + 31) / 32, HH / 32);
        mh_transpose<<<tg, tb, 0, stream>>>(hv, T);

        // 2) coalesced row-gather + min, L2-resident table
        dim3 gg(BB, NCHUNK);
        mh_gather_min<<<gg, 192, 0, stream>>>(ids, mask, T, out);
    } else {
        // 2b) fallback: strided direct gather
        dim3 gg(BB, NCHUNK);
        mh_gather_min_direct<<<gg, 256, 0, stream>>>(ids, mask, hv, out);
    }

    // 3) WMMA Gram-diagonal row norms + in-place scale
    mh_normalize_wmma<<<BB / 16, 32, 0, stream>>>(out);
}